// UMA_21852793602399
// MI455X (gfx1250) — compile-verified
//
#include <hip/hip_runtime.h>

#define B_  32
#define L_  3000
#define D_  256
#define TPR 188            // tiles of 16 frames per row: ceil(3000/16)
#define LP_ (TPR * 16)     // padded score row stride = 3008
#define LDSTRIDE 260       // 256 + 4-float pad -> bank stride 4, conflict-free
#define WPB 2              // waves per block in kernel 1

typedef __attribute__((ext_vector_type(2))) float v2f;
typedef __attribute__((ext_vector_type(4))) float v4f;
typedef __attribute__((ext_vector_type(8))) float v8f;
typedef __attribute__((ext_vector_type(4))) int   v4i;

typedef __attribute__((address_space(1))) v4i gas_v4i;   // global (AS1) 16B chunk
typedef __attribute__((address_space(3))) v4i las_v4i;   // LDS (AS3) 16B chunk

// ---- async global->LDS copy (16B per lane), with compile-safe fallback ----
#if __has_builtin(__builtin_amdgcn_global_load_async_to_lds_b128)
#define HAS_ASYNC_LDS 1
__device__ __forceinline__ void async_copy16(const float* g, float* l) {
  __builtin_amdgcn_global_load_async_to_lds_b128(
      (gas_v4i*)(g), (las_v4i*)(l), 0, 0);
}
__device__ __forceinline__ void async_wait0() {
#if __has_builtin(__builtin_amdgcn_s_wait_asynccnt)
  __builtin_amdgcn_s_wait_asynccnt(0);
#else
  asm volatile("s_wait_asynccnt 0" ::: "memory");
#endif
}
#else
#define HAS_ASYNC_LDS 0
__device__ __forceinline__ void async_copy16(const float* g, float* l) {
  *(v4f*)l = *(const v4f*)g;
}
__device__ __forceinline__ void async_wait0() {}
#endif

__device__ __forceinline__ float fast_sigmoid(float x) {
  return __builtin_amdgcn_rcpf(1.0f + __expf(-x));   // v_rcp_f32 + v_exp_f32
}

// ---------------------------------------------------------------------------
// Kernel 1: scalar[b,l] = sigmoid(xs[b,l,:] . W + bias) via V_WMMA_F32_16X16X4_F32.
// Each wave: async-stage its 16x256 tile (+W) into a private LDS region,
// s_wait_asynccnt, then 64 chained WMMAs fed by ds_load_b64.
// B-matrix = W broadcast into all 16 columns -> D[m,n] == score[m] for all n;
// lanes 0/16 hold the 16 scores in their 8 accumulator VGPRs.
// ---------------------------------------------------------------------------
__global__ void __launch_bounds__(WPB * 32)
score_wmma_kernel(const float* __restrict__ xs,
                  const float* __restrict__ W,
                  const float* __restrict__ bias,
                  float* __restrict__ sarr) {
  __shared__ float lds[WPB * (16 * LDSTRIDE + D_)];
  const int wslot = (int)(threadIdx.x >> 5);
  const int lane  = (int)(threadIdx.x & 31);
  const int wave  = (int)blockIdx.x * WPB + wslot;
  const int totalTiles = B_ * TPR;
  if (wave >= totalTiles) return;               // wave-uniform

  const int b  = wave / TPR;
  const int m0 = (wave % TPR) * 16;
  float* myLds = lds + wslot * (16 * LDSTRIDE + D_);
  float* wLdsBase = myLds + 16 * LDSTRIDE;

  // ---- stage 16 rows x 1KB (1024 x 16B chunks) + W (64 x 16B) into LDS ----
  const float* xbase = xs + (size_t)b * L_ * D_;
#pragma unroll 4
  for (int i = 0; i < 32; ++i) {
    const int c   = i * 32 + lane;              // chunk id 0..1023
    const int row = c >> 6;
    const int col = (c & 63) << 2;              // float index within row
    int mr = m0 + row; if (mr > L_ - 1) mr = L_ - 1;   // clamp tail tile
    async_copy16(xbase + (size_t)mr * D_ + col, myLds + row * LDSTRIDE + col);
  }
#pragma unroll
  for (int i = lane; i < 64; i += 32)
    async_copy16(W + (i << 2), wLdsBase + (i << 2));
  async_wait0();

  // ---- 64 chained f32 WMMAs over K=256 ----
  const int koff = (lane >> 4) << 1;            // lanes 0-15 -> K+0/1, 16-31 -> K+2/3
  const float* aLds = myLds + (lane & 15) * LDSTRIDE + koff;
  const float* wLds = wLdsBase + koff;

  v8f c = {0.f, 0.f, 0.f, 0.f, 0.f, 0.f, 0.f, 0.f};
#pragma unroll 4
  for (int k = 0; k < D_; k += 4) {
    v2f a  = *(const v2f*)(aLds + k);           // ds_load_b64
    v2f w2 = *(const v2f*)(wLds + k);           // ds_load_b64 (broadcast)
    c = __builtin_amdgcn_wmma_f32_16x16x4_f32(false, a, false, w2,
                                              (short)0, c, false, false);
  }

  // ---- lanes 0/16 write 8 contiguous sigmoids each (padded row, no guards) --
  const float bias0 = bias[0];
  v4f o0, o1;
#pragma unroll
  for (int r = 0; r < 4; ++r) o0[r] = fast_sigmoid(c[r] + bias0);
#pragma unroll
  for (int r = 0; r < 4; ++r) o1[r] = fast_sigmoid(c[r + 4] + bias0);
  if ((lane & 15) == 0) {
    const int fbase = m0 + ((lane >> 4) << 3);  // lane0 -> m0, lane16 -> m0+8
    float* sp = sarr + (size_t)b * LP_ + fbase;
    *(v4f*)(sp)     = o0;
    *(v4f*)(sp + 4) = o1;
  }
}

// ---------------------------------------------------------------------------
// Kernel 2: valley-boundary extraction + stream compaction.
// One wave32 per batch row; ballot+popcount compaction, separate start/end
// running counters (t==0 forces a start, t==L-1 forces an end).
// ---------------------------------------------------------------------------
__global__ void valley_kernel(const float* __restrict__ sarr,
                              int* __restrict__ starts,
                              int* __restrict__ ends,
                              int* __restrict__ counts,
                              int Kmax) {
  const int b    = (int)blockIdx.x;
  const int lane = (int)threadIdx.x;            // 32 threads
  const float* sr = sarr + (size_t)b * LP_;
  int ns = 0, ne = 0;
  const unsigned ltmask = (lane == 0) ? 0u : (~0u >> (32 - lane));

  for (int t0 = 0; t0 < L_; t0 += 32) {
    const int  t   = t0 + lane;
    const bool inb = (t < L_);
    const float sc = inb ? sr[t] : 1.0f;
    const float sb = (inb && t > 0)      ? sr[t - 1] : 0.0f;  // "before" pads 0
    const float sa = (inb && t < L_ - 1) ? sr[t + 1] : 0.0f;  // "after"  pads 0
    const bool valley = inb && (sc < sb) && (sc < sa);
    const bool startm = inb && (valley || (t == 0));
    const bool endm   = inb && ((valley && t > 0) || (t == L_ - 1));

    const unsigned smask = __builtin_amdgcn_ballot_w32(startm);
    const unsigned emask = __builtin_amdgcn_ballot_w32(endm);
    if (startm) {
      const int idx = ns + __builtin_popcount(smask & ltmask);
      if (idx < Kmax) starts[(size_t)b * Kmax + idx] = t;
    }
    if (endm) {
      const int idx = ne + __builtin_popcount(emask & ltmask);
      int e = t + 2; if (e > L_) e = L_;
      if (idx < Kmax) ends[(size_t)b * Kmax + idx] = e;
    }
    ns += __builtin_popcount(smask);
    ne += __builtin_popcount(emask);
  }
  // pad unused segments with [0,0) -> zero output rows (matches torch zero-init)
  for (int k = ns + lane; k < Kmax; k += 32) {
    starts[(size_t)b * Kmax + k] = 0;
    ends[(size_t)b * Kmax + k]   = 0;
  }
  if (lane == 0) counts[b] = ns;
}

// ---------------------------------------------------------------------------
// Kernel 3: out[b,k,:] = (sum_{t in [s,e)} scalar[t]*x[b,t,:]) / max(sum scalar, 1e-6)
// One 64-thread block per (k, b); each thread owns a float4 slice of D=256.
// ---------------------------------------------------------------------------
__global__ void segsum_kernel(const float* __restrict__ xs,
                              const float* __restrict__ sarr,
                              const int* __restrict__ starts,
                              const int* __restrict__ ends,
                              const int* __restrict__ counts,
                              float* __restrict__ out,
                              int Kmax) {
  const int b   = (int)blockIdx.y;
  const int k   = (int)blockIdx.x;
  const int tid = (int)threadIdx.x;             // 64 threads * float4 = 256
  v4f   acc  = {0.f, 0.f, 0.f, 0.f};
  float ssum = 0.f;

  if (k < counts[b]) {                          // block-uniform branch
    const int st = starts[(size_t)b * Kmax + k];
    const int en = ends[(size_t)b * Kmax + k];
    const float* xb = xs + (size_t)b * L_ * D_ + tid * 4;
    const float* sp = sarr + (size_t)b * LP_;
    for (int t = st; t < en; ++t) {
      __builtin_prefetch(xb + (size_t)(t + 1) * D_, 0, 0);   // global_prefetch_b8
      const float sc = sp[t];
      const v4f x4 = *(const v4f*)(xb + (size_t)t * D_);
      acc += sc * x4;
      ssum += sc;
    }
  }
  const float inv = 1.0f / fmaxf(ssum, 1e-6f);
  const v4f o = acc * inv;
  *(v4f*)(out + ((size_t)b * Kmax + k) * D_ + tid * 4) = o;
}

// ---------------------------------------------------------------------------
// Kernel 4: counts -> tail of d_out (as float values)
// ---------------------------------------------------------------------------
__global__ void counts_kernel(const int* __restrict__ counts,
                              float* __restrict__ out, int off) {
  const int b = (int)threadIdx.x;
  if (b < B_) out[off + b] = (float)counts[b];
}

extern "C" void kernel_launch(void* const* d_in, const int* in_sizes, int n_in,
                              void* d_out, int out_size, void* d_ws, size_t ws_size,
                              hipStream_t stream) {
  (void)in_sizes; (void)n_in; (void)ws_size;
  const float* xs   = (const float*)d_in[0];
  // d_in[1] = olens : unused by the reference math
  const float* W    = (const float*)d_in[2];
  const float* bias = (const float*)d_in[3];
  float* out = (float*)d_out;

  // out_size = B*K*D + B  ->  recover the data-dependent K
  int Kmax = (out_size - B_) / (B_ * D_);
  if (Kmax < 1) Kmax = 1;

  // workspace layout
  char* ws = (char*)d_ws;
  float* sarr = (float*)ws;                     // B * LP_ floats (padded rows)
  size_t off = (size_t)B_ * LP_ * sizeof(float);
  off = (off + 255) & ~(size_t)255;
  int* starts = (int*)(ws + off); off += (size_t)B_ * Kmax * sizeof(int);
  off = (off + 255) & ~(size_t)255;
  int* ends   = (int*)(ws + off); off += (size_t)B_ * Kmax * sizeof(int);
  off = (off + 255) & ~(size_t)255;
  int* counts = (int*)(ws + off);

  const int totalTiles = B_ * TPR;              // one wave per 16-frame tile
  const int blocks = (totalTiles + WPB - 1) / WPB;
  score_wmma_kernel<<<blocks, WPB * 32, 0, stream>>>(xs, W, bias, sarr);

  valley_kernel<<<B_, 32, 0, stream>>>(sarr, starts, ends, counts, Kmax);

  dim3 grid((unsigned)Kmax, B_);
  segsum_kernel<<<grid, 64, 0, stream>>>(xs, sarr, starts, ends, counts, out, Kmax);

  counts_kernel<<<1, 32, 0, stream>>>(counts, out, B_ * Kmax * D_);
}